// Interaction_36644660969764
// MI455X (gfx1250) — compile-verified
//
#include <hip/hip_runtime.h>
#include <math.h>

// ---------------------------------------------------------------------------
// Types for CDNA5 WMMA f32 16x16x4:  A,B = 2 VGPRs/lane (v2f), C/D = 8 (v8f)
// ---------------------------------------------------------------------------
typedef __attribute__((ext_vector_type(2))) float v2f;
typedef __attribute__((ext_vector_type(8))) float v8f;

#define EMBC      64      // channels
#define NPB       16      // nodes per block (node kernel)
#define GS        68      // LDS row stride (64 + 4 pad) -> conflict-free b64
#define NCOLS     144     // 16 (I) + 48 (A) + 80 (S) columns per node tile
#define NODE_BLK  128     // 4 waves
#define EDGE_BLK  128     // 4 waves
#define EPB       64      // edges per block
#define RADN      32
#define WDS       36      // 32 + 4 pad stride for W_dist / phi in LDS

__device__ __forceinline__ v8f wmma_f32_k4(v2f a, v2f b, v8f c) {
  // D = A(16x4 f32) * B(4x16 f32) + C(16x16 f32)
  return __builtin_amdgcn_wmma_f32_16x16x4_f32(
      /*neg_a=*/false, a, /*neg_b=*/false, b,
      /*c_mod=*/(short)0, c, /*reuse_a=*/false, /*reuse_b=*/false);
}

// ---------------------------------------------------------------------------
// Kernel 0: zero the per-node coefficient accumulator F[N][192]
// ---------------------------------------------------------------------------
__global__ void zero_kernel(float* __restrict__ p, int n) {
  int i = blockIdx.x * blockDim.x + threadIdx.x;
  if (i < n) p[i] = 0.0f;
}

// ---------------------------------------------------------------------------
// Kernel 1: per-edge RBF coefficients via WMMA GEMM, scatter-add into F.
//   phi  : (E x 32)      A-operand (16 edges x K4 per wmma)
//   WdT  : (32 x 192)    B-operand, stored as W_dist[j][k] with padded stride
//   coeff[e][j] = (sum_k phi[e][k]*W_dist[j][k] + b[j]) * env(d_e)
//   F[nb[e]][j] += coeff[e][j]   (global f32 atomics, L2-resident table)
// ---------------------------------------------------------------------------
__global__ __launch_bounds__(EDGE_BLK)
void edge_coeff_kernel(const float* __restrict__ dist,
                       const int*   __restrict__ nbr,
                       const float* __restrict__ Wdist,   // (192,32) row-major
                       const float* __restrict__ bdist,   // (192,)
                       float*       __restrict__ F,       // (N,192)
                       int E) {
  __shared__ float Wd[192 * WDS];
  __shared__ float phiL[EPB * WDS];
  __shared__ float bL[192];
  __shared__ float dL[EPB];
  __shared__ float envL[EPB];
  __shared__ int   nbL[EPB];

  const int tid   = threadIdx.x;
  const int ebase = blockIdx.x * EPB;

  // RBF constants (PhysNet exp-normal), CUTOFF = 5
  const float E5   = 0.00673794699909f;          // exp(-5)
  const float MU0  = E5;
  const float DMU  = (1.0f - E5) / 31.0f;
  const float ISTP = 16.0f / (1.0f - E5);        // 1 / ((2/32)*(1-e5))
  const float BETA = ISTP * ISTP;
  const float PI_F = 3.14159265358979f;

  for (int i = tid; i < 192 * RADN; i += EDGE_BLK) {
    int j = i >> 5, k = i & 31;
    Wd[j * WDS + k] = Wdist[i];
  }
  for (int i = tid; i < 192; i += EDGE_BLK) bL[i] = bdist[i];
  for (int i = tid; i < EPB; i += EDGE_BLK) {
    int e = ebase + i;
    float d = (e < E) ? dist[e] : 1.0e9f;
    dL[i]   = d;
    envL[i] = (d < 5.0f) ? 0.5f * (__cosf(PI_F * d * 0.2f) + 1.0f) : 0.0f;
    nbL[i]  = (e < E) ? nbr[e] : 0;
  }
  __syncthreads();

  for (int i = tid; i < EPB * RADN; i += EDGE_BLK) {
    int el = i >> 5, k = i & 31;
    float ed = __expf(-dL[el]);
    float t  = ed - (MU0 + (float)k * DMU);
    phiL[el * WDS + k] = __expf(-BETA * t * t);
  }
  __syncthreads();

  const int  wave = tid >> 5;
  const int  lane = tid & 31;
  const int  r    = lane & 15;
  const bool hi   = lane >= 16;
  const int  koff = hi ? 2 : 0;
  const int  osel = hi ? 8 : 0;

  // A operand: rows = 16 edges of this wave, ISA layout (VGPR0: K0/K2, VGPR1: K1/K3)
  v2f pa[8];
#pragma unroll
  for (int kk = 0; kk < 8; ++kk)
    pa[kk] = *reinterpret_cast<const v2f*>(phiL + (wave * 16 + r) * WDS + kk * 4 + koff);

  for (int jt = 0; jt < 12; ++jt) {
    v8f acc = {0.f, 0.f, 0.f, 0.f, 0.f, 0.f, 0.f, 0.f};
#pragma unroll
    for (int kk = 0; kk < 8; ++kk) {
      v2f b = *reinterpret_cast<const v2f*>(Wd + (jt * 16 + r) * WDS + kk * 4 + koff);
      acc = wmma_f32_k4(pa[kk], b, acc);
    }
    int   j  = jt * 16 + r;          // D col = lane%16
    float bj = bL[j];
#pragma unroll
    for (int v = 0; v < 8; ++v) {
      int   el  = wave * 16 + osel + v;   // D row = v + 8*hi
      float val = (acc[v] + bj) * envL[el];
      atomicAdd(F + (size_t)nbL[el] * 192 + j, val);
    }
  }
}

// ---------------------------------------------------------------------------
// Channel-mix: H[col][o] = sum_c W[o][c] * G[col][c] on the compressed 9-dof
// basis. 4 waves x (1 I-tile + 3 A-tiles + 5 S-tiles) x 16 K-steps of f32 WMMA.
// ---------------------------------------------------------------------------
__device__ __forceinline__ void mix_group(const float* __restrict__ W,
                                          int ntiles, int colbase,
                                          const float* Gs, float* Hs,
                                          int lane, int obase) {
  const int  r    = lane & 15;
  const bool hi   = lane >= 16;
  const int  koff = hi ? 2 : 0;
  const int  osel = hi ? 8 : 0;

  v2f wa[16];                      // W rows o=obase+r, all 64 K values
#pragma unroll
  for (int kk = 0; kk < 16; ++kk)
    wa[kk] = *reinterpret_cast<const v2f*>(W + (obase + r) * EMBC + kk * 4 + koff);

  for (int t = 0; t < ntiles; ++t) {
    const int col = colbase + t * 16 + r;
    v8f acc = {0.f, 0.f, 0.f, 0.f, 0.f, 0.f, 0.f, 0.f};
#pragma unroll
    for (int kk = 0; kk < 16; ++kk) {
      v2f b = *reinterpret_cast<const v2f*>(Gs + col * GS + kk * 4 + koff);
      acc = wmma_f32_k4(wa[kk], b, acc);
    }
#pragma unroll
    for (int v = 0; v < 8; ++v)
      Hs[col * GS + obase + osel + v] = acc[v];
  }
}

__device__ __forceinline__ void mm3(const float a[9], const float b[9], float c[9]) {
#pragma unroll
  for (int i = 0; i < 3; ++i)
#pragma unroll
    for (int j = 0; j < 3; ++j)
      c[i * 3 + j] = a[i * 3 + 0] * b[0 + j] + a[i * 3 + 1] * b[3 + j] + a[i * 3 + 2] * b[6 + j];
}

// ---------------------------------------------------------------------------
// Kernel 2: per 16-node tile: normalize+decompose -> WMMA pre-mix ->
//           Z = Y M + M Y (M rebuilt from scattered F) -> renorm+decompose ->
//           WMMA post-mix -> out = Y + Y@Y
// Column map per local node t: I -> t ; A_p -> 16+16p+t ; S_q -> 64+16q+t
// ---------------------------------------------------------------------------
__global__ __launch_bounds__(NODE_BLK)
void node_kernel(const float* __restrict__ X,
                 const float* __restrict__ F,
                 const float* __restrict__ WIpre, const float* __restrict__ WApre,
                 const float* __restrict__ WSpre,
                 const float* __restrict__ WIpost, const float* __restrict__ WApost,
                 const float* __restrict__ WSpost,
                 float* __restrict__ out, int N) {
  __shared__ float G[NCOLS * GS];   // decomposition (K = input channel)
  __shared__ float H[NCOLS * GS];   // mixed result  (index = output channel)

  const int tid = threadIdx.x;
  const int nb0 = blockIdx.x * NPB;
  const int lane = tid & 31;
  const int wave = tid >> 5;

  // ---- phase 1: normalize + decompose -> G -------------------------------
  for (int t = tid; t < NPB * EMBC; t += NODE_BLK) {
    const int nl = t >> 6, ch = t & 63, n = nb0 + nl;
    float x[9];
    if (n < N) {
      const float* xp = X + ((size_t)n * EMBC + ch) * 9;
      float frob = 0.f;
#pragma unroll
      for (int i = 0; i < 9; ++i) { x[i] = xp[i]; frob += x[i] * x[i]; }
      const float inv = 1.0f / (frob + 1.0f);
#pragma unroll
      for (int i = 0; i < 9; ++i) x[i] *= inv;
    } else {
#pragma unroll
      for (int i = 0; i < 9; ++i) x[i] = 0.f;
    }
    const float m = (x[0] + x[4] + x[8]) * (1.0f / 3.0f);
    G[(nl)       * GS + ch] = m;
    G[(16 + nl)  * GS + ch] = 0.5f * (x[1] - x[3]);
    G[(32 + nl)  * GS + ch] = 0.5f * (x[2] - x[6]);
    G[(48 + nl)  * GS + ch] = 0.5f * (x[5] - x[7]);
    G[(64 + nl)  * GS + ch] = x[0] - m;
    G[(80 + nl)  * GS + ch] = x[4] - m;
    G[(96 + nl)  * GS + ch] = 0.5f * (x[1] + x[3]);
    G[(112 + nl) * GS + ch] = 0.5f * (x[2] + x[6]);
    G[(128 + nl) * GS + ch] = 0.5f * (x[5] + x[7]);
  }
  __syncthreads();

  // ---- phase 2: WMMA pre-mix ---------------------------------------------
  mix_group(WIpre, 1, 0,  G, H, lane, wave * 16);
  mix_group(WApre, 3, 16, G, H, lane, wave * 16);
  mix_group(WSpre, 5, 64, G, H, lane, wave * 16);
  __syncthreads();

  // ---- phase 3: Z = Y M + M Y, renormalize, re-decompose -> G ------------
  for (int t = tid; t < NPB * EMBC; t += NODE_BLK) {
    const int nl = t >> 6, ch = t & 63, n = nb0 + nl;
    if (n >= N) continue;

    // rebuild Y[n][ch] from mixed irreps in H
    const float ym  = H[(nl)       * GS + ch];
    const float ya0 = H[(16 + nl)  * GS + ch];
    const float ya1 = H[(32 + nl)  * GS + ch];
    const float ya2 = H[(48 + nl)  * GS + ch];
    const float ys0 = H[(64 + nl)  * GS + ch];
    const float ys1 = H[(80 + nl)  * GS + ch];
    const float ys2 = H[(96 + nl)  * GS + ch];
    const float ys3 = H[(112 + nl) * GS + ch];
    const float ys4 = H[(128 + nl) * GS + ch];
    float Y[9] = { ym + ys0,  ya0 + ys2,      ya1 + ys3,
                  -ya0 + ys2, ym + ys1,       ya2 + ys4,
                  -ya1 + ys3, -ya2 + ys4,     ym - ys0 - ys1 };

    // rebuild M[n][ch] from original decomposition (G) and scattered coeffs F
    const float m  = G[(nl)       * GS + ch];
    const float a0 = G[(16 + nl)  * GS + ch];
    const float a1 = G[(32 + nl)  * GS + ch];
    const float a2 = G[(48 + nl)  * GS + ch];
    const float s0 = G[(64 + nl)  * GS + ch];
    const float s1 = G[(80 + nl)  * GS + ch];
    const float s2 = G[(96 + nl)  * GS + ch];
    const float s3 = G[(112 + nl) * GS + ch];
    const float s4 = G[(128 + nl) * GS + ch];
    const float fI = F[(size_t)n * 192 + ch];
    const float fA = F[(size_t)n * 192 + 64 + ch];
    const float fS = F[(size_t)n * 192 + 128 + ch];
    float M[9] = { fI * m + fS * s0,  fA * a0 + fS * s2,  fA * a1 + fS * s3,
                  -fA * a0 + fS * s2, fI * m + fS * s1,   fA * a2 + fS * s4,
                  -fA * a1 + fS * s3, -fA * a2 + fS * s4, fI * m - fS * (s0 + s1) };

    float t1[9], t2[9], Z[9];
    mm3(Y, M, t1);
    mm3(M, Y, t2);
    float nrm = 0.f;
#pragma unroll
    for (int i = 0; i < 9; ++i) {
      Z[i] = t1[i] + t2[i];
      const float u = Z[i] + 1.0f;
      nrm += u * u;
    }
    const float invn = 1.0f / nrm;
#pragma unroll
    for (int i = 0; i < 9; ++i) Z[i] *= invn;

    const float m2 = (Z[0] + Z[4] + Z[8]) * (1.0f / 3.0f);
    G[(nl)       * GS + ch] = m2;
    G[(16 + nl)  * GS + ch] = 0.5f * (Z[1] - Z[3]);
    G[(32 + nl)  * GS + ch] = 0.5f * (Z[2] - Z[6]);
    G[(48 + nl)  * GS + ch] = 0.5f * (Z[5] - Z[7]);
    G[(64 + nl)  * GS + ch] = Z[0] - m2;
    G[(80 + nl)  * GS + ch] = Z[4] - m2;
    G[(96 + nl)  * GS + ch] = 0.5f * (Z[1] + Z[3]);
    G[(112 + nl) * GS + ch] = 0.5f * (Z[2] + Z[6]);
    G[(128 + nl) * GS + ch] = 0.5f * (Z[5] + Z[7]);
  }
  __syncthreads();

  // ---- phase 4: WMMA post-mix --------------------------------------------
  mix_group(WIpost, 1, 0,  G, H, lane, wave * 16);
  mix_group(WApost, 3, 16, G, H, lane, wave * 16);
  mix_group(WSpost, 5, 64, G, H, lane, wave * 16);
  __syncthreads();

  // ---- phase 5: out = Y + Y @ Y ------------------------------------------
  for (int t = tid; t < NPB * EMBC; t += NODE_BLK) {
    const int nl = t >> 6, ch = t & 63, n = nb0 + nl;
    if (n >= N) continue;
    const float ym  = H[(nl)       * GS + ch];
    const float ya0 = H[(16 + nl)  * GS + ch];
    const float ya1 = H[(32 + nl)  * GS + ch];
    const float ya2 = H[(48 + nl)  * GS + ch];
    const float ys0 = H[(64 + nl)  * GS + ch];
    const float ys1 = H[(80 + nl)  * GS + ch];
    const float ys2 = H[(96 + nl)  * GS + ch];
    const float ys3 = H[(112 + nl) * GS + ch];
    const float ys4 = H[(128 + nl) * GS + ch];
    float P[9] = { ym + ys0,  ya0 + ys2,      ya1 + ys3,
                  -ya0 + ys2, ym + ys1,       ya2 + ys4,
                  -ya1 + ys3, -ya2 + ys4,     ym - ys0 - ys1 };
    float Q[9];
    mm3(P, P, Q);
    float* op = out + ((size_t)n * EMBC + ch) * 9;
#pragma unroll
    for (int i = 0; i < 9; ++i) op[i] = P[i] + Q[i];
  }
}

// ---------------------------------------------------------------------------
extern "C" void kernel_launch(void* const* d_in, const int* in_sizes, int n_in,
                              void* d_out, int out_size, void* d_ws, size_t ws_size,
                              hipStream_t stream) {
  const float* X      = (const float*)d_in[0];
  const float* dist   = (const float*)d_in[1];
  const int*   nbr    = (const int*)  d_in[2];
  const float* WIpre  = (const float*)d_in[3];
  const float* WApre  = (const float*)d_in[4];
  const float* WSpre  = (const float*)d_in[5];
  const float* WIpost = (const float*)d_in[6];
  const float* WApost = (const float*)d_in[7];
  const float* WSpost = (const float*)d_in[8];
  const float* Wdist  = (const float*)d_in[9];
  const float* bdist  = (const float*)d_in[10];
  float*       out    = (float*)d_out;

  const int N = in_sizes[0] / (EMBC * 9);   // 20000
  const int E = in_sizes[1];                // 320000

  float* F = (float*)d_ws;                  // N x 192 accumulator (15.4 MB)
  const int fcount = N * 192;

  zero_kernel<<<(fcount + 255) / 256, 256, 0, stream>>>(F, fcount);
  edge_coeff_kernel<<<(E + EPB - 1) / EPB, EDGE_BLK, 0, stream>>>(
      dist, nbr, Wdist, bdist, F, E);
  node_kernel<<<(N + NPB - 1) / NPB, NODE_BLK, 0, stream>>>(
      X, F, WIpre, WApre, WSpre, WIpost, WApost, WSpost, out, N);
}